// _GradientBoostingBase_63282048139716
// MI455X (gfx1250) — compile-verified
//
#include <hip/hip_runtime.h>
#include <stdint.h>

#define N_STAGES   400
#define N_CLASSES  10
#define N_TREES    (N_STAGES * N_CLASSES)   // 4000
#define DEPTH      6
#define N_INTERNAL 63
#define N_LEAVES   64
#define N_FEATURES 128
#define BATCH      32768
#define LR         0.1f

#define TILE_S     128        // samples per block == threads per block
#define BLOCK      128
#define XSTR       132        // padded floats per sample row (16B-aligned rows)
#define TC         40         // trees per LDS chunk (multiple of N_CLASSES)
#define SPLITS     4
#define TREES_PER_SPLIT (N_TREES / SPLITS)  // 1000 (multiple of TC and of 10)

// ---------------- CDNA5 async global->LDS copy primitives ----------------
// GVS form: lds_byte_addr (VGPR), 32-bit voffset (VGPR), 64-bit base (SGPR).
// Tracked by ASYNCcnt; completion enforced with s_wait_asynccnt.
__device__ __forceinline__ void async_cp16(uint32_t lds_addr, uint64_t gbase,
                                           uint32_t voff) {
#if defined(__gfx1250__)
  asm volatile("global_load_async_to_lds_b128 %0, %1, %2"
               :: "v"(lds_addr), "v"(voff), "s"(gbase)
               : "memory");
#else
  (void)lds_addr; (void)gbase; (void)voff;
#endif
}

__device__ __forceinline__ void wait_async0() {
#if defined(__gfx1250__)
  asm volatile("s_wait_asynccnt 0x0" ::: "memory");
#endif
}

__device__ __forceinline__ uint32_t lds_off(const void* p) {
  // Generic LDS address = aperture(high 32) | byte offset(low 32).
  return (uint32_t)(uintptr_t)p;
}

// ------------------- pack kernel: interleave {feat, thr} -----------------
__global__ __launch_bounds__(256) void gbt_pack(
    const int* __restrict__ features, const float* __restrict__ thresholds,
    int2* __restrict__ packed) {
  const int i = blockIdx.x * blockDim.x + threadIdx.x;
  if (i < N_TREES * N_INTERNAL)
    packed[i] = make_int2(features[i], __float_as_int(thresholds[i]));
}

// -------------------------------- main kernel ----------------------------
template <bool PACKED>
__global__ __launch_bounds__(BLOCK) void gbt_kernel(
    const float* __restrict__ x, const int* __restrict__ features,
    const float* __restrict__ thresholds, const int2* __restrict__ packed,
    const float* __restrict__ leaves, const float* __restrict__ init_out,
    float* __restrict__ dst_base, int trees_per_split, int apply_init) {
  __shared__ __align__(16) float s_x[TILE_S * XSTR];                 // 67584 B
  // PACKED:   int2 nodes [TC*63]          (20160 B per buffer)
  // UNPACKED: int feat [TC*63] then float thr [TC*63] (same 20160 B)
  __shared__ __align__(16) char  s_node[2][TC * N_INTERNAL * 8];     // 40320 B
  __shared__ __align__(16) float s_leaf[2][TC * N_LEAVES];           // 20480 B

  const int tid = threadIdx.x;
  const int s0  = blockIdx.x * TILE_S;
  const int tree_base = blockIdx.y * trees_per_split;                // %10 == 0
  const int n_chunks  = trees_per_split / TC;

  // ---- stage x tile: 128 samples x 32 x 16B chunks, async into padded rows
  {
    const uint64_t gx = (uint64_t)x + (uint64_t)s0 * N_FEATURES * 4u;
    const uint32_t lx = lds_off(&s_x[0]);
    constexpr int XCH = TILE_S * (N_FEATURES * 4 / 16);              // 4096
    for (int i = tid; i < XCH; i += BLOCK) {
      int s = i >> 5, c = i & 31;
      async_cp16(lx + (uint32_t)(s * XSTR * 4 + c * 16),
                 gx, (uint32_t)(s * N_FEATURES * 4 + c * 16));
    }
  }

  // ---- chunk staging (async, 16B granules) ----
  auto stage_chunk = [&](int buf, int t0) {
    const uint32_t ln = lds_off(&s_node[buf][0]);
    const uint32_t ll = lds_off(&s_leaf[buf][0]);
    const uint64_t gl = (uint64_t)(leaves + (size_t)t0 * N_LEAVES);
    if constexpr (PACKED) {
      const uint64_t gn = (uint64_t)(packed + (size_t)t0 * N_INTERNAL);
      constexpr int NODE16 = (TC * N_INTERNAL * 8) / 16;             // 1260
      for (int i = tid; i < NODE16; i += BLOCK)
        async_cp16(ln + (uint32_t)i * 16u, gn, (uint32_t)i * 16u);
    } else {
      const uint64_t gf = (uint64_t)(features   + (size_t)t0 * N_INTERNAL);
      const uint64_t gt = (uint64_t)(thresholds + (size_t)t0 * N_INTERNAL);
      constexpr int HALF = TC * N_INTERNAL * 4;                      // 10080 B
      constexpr int NODE16 = HALF / 16;                              // 630
      for (int i = tid; i < NODE16; i += BLOCK) {
        uint32_t b = (uint32_t)i * 16u;
        async_cp16(ln + b, gf, b);
        async_cp16(ln + HALF + b, gt, b);
      }
    }
    constexpr int LEAF16 = (TC * N_LEAVES * 4) / 16;                 // 640
    for (int i = tid; i < LEAF16; i += BLOCK)
      async_cp16(ll + (uint32_t)i * 16u, gl, (uint32_t)i * 16u);
  };

  stage_chunk(0, tree_base);
  wait_async0();
  __syncthreads();

  float acc[N_CLASSES];
#pragma unroll
  for (int k = 0; k < N_CLASSES; ++k) acc[k] = 0.0f;

  const float* __restrict__ myx = &s_x[tid * XSTR];

  for (int c = 0; c < n_chunks; ++c) {
    const int buf = c & 1;
    if (c + 1 < n_chunks) stage_chunk(buf ^ 1, tree_base + (c + 1) * TC);

    const int2*  __restrict__ lnode = (const int2*)&s_node[buf][0];
    const int*   __restrict__ lfeat = (const int*)&s_node[buf][0];
    const float* __restrict__ lthr =
        (const float*)&s_node[buf][TC * N_INTERNAL * 4];
    const float* __restrict__ lleaf = &s_leaf[buf][0];

    for (int g = 0; g < TC / N_CLASSES; ++g) {   // 4 groups of 10 trees
#pragma unroll
      for (int k = 0; k < N_CLASSES; ++k) {      // class index is compile-time
        const int tb = (g * N_CLASSES + k) * N_INTERNAL;
        int idx = 0;
#pragma unroll
        for (int l = 0; l < DEPTH; ++l) {
          int f; float thv;
          if constexpr (PACKED) {
            const int2 nd = lnode[tb + idx];     // one ds_load_b64
            f = nd.x;
            thv = __int_as_float(nd.y);
          } else {
            f = lfeat[tb + idx];
            thv = lthr[tb + idx];
          }
          const float xv = myx[f];
          idx = 2 * idx + 1 + (xv > thv ? 1 : 0);
        }
        acc[k] += lleaf[(g * N_CLASSES + k) * N_LEAVES + (idx - N_INTERNAL)];
      }
    }
    wait_async0();      // next chunk's async copies have landed by now
    __syncthreads();    // publish to all waves before overwriting/reading
  }

  // ---- write result: partial (raw sums) or final (init + LR * sum)
  float* __restrict__ dst = dst_base + (size_t)blockIdx.y * BATCH * N_CLASSES;
  const int n = s0 + tid;
  if (apply_init) {
#pragma unroll
    for (int k = 0; k < N_CLASSES; ++k)
      dst[(size_t)n * N_CLASSES + k] = init_out[k] + LR * acc[k];
  } else {
#pragma unroll
    for (int k = 0; k < N_CLASSES; ++k)
      dst[(size_t)n * N_CLASSES + k] = acc[k];
  }
}

// ------------------------- deterministic reduction -----------------------
__global__ __launch_bounds__(256) void gbt_reduce(
    const float* __restrict__ partials, const float* __restrict__ init_out,
    float* __restrict__ out) {
  const int i = blockIdx.x * blockDim.x + threadIdx.x;
  if (i >= BATCH * N_CLASSES) return;
  const int k = i % N_CLASSES;
  float s = 0.0f;
#pragma unroll
  for (int sp = 0; sp < SPLITS; ++sp)
    s += partials[(size_t)sp * BATCH * N_CLASSES + i];
  out[i] = init_out[k] + LR * s;
}

// -------------------------------- launcher -------------------------------
extern "C" void kernel_launch(void* const* d_in, const int* in_sizes, int n_in,
                              void* d_out, int out_size, void* d_ws,
                              size_t ws_size, hipStream_t stream) {
  const float* x          = (const float*)d_in[0];
  const int*   features   = (const int*)  d_in[1];
  const float* thresholds = (const float*)d_in[2];
  const float* leaves     = (const float*)d_in[3];
  const float* init_out   = (const float*)d_in[4];
  float*       out        = (float*)d_out;

  const size_t packed_bytes  = (size_t)N_TREES * N_INTERNAL * 8;   // 2016000
  const size_t partial_bytes =
      (size_t)SPLITS * BATCH * N_CLASSES * sizeof(float);          // 5242880
  const int pack_elems = N_TREES * N_INTERNAL;

  if (ws_size >= packed_bytes + partial_bytes) {
    // Best path: packed nodes + 4-way tree split + deterministic reduce.
    int2*  packed   = (int2*)d_ws;
    float* partials = (float*)((char*)d_ws + packed_bytes);
    gbt_pack<<<(pack_elems + 255) / 256, 256, 0, stream>>>(features,
                                                           thresholds, packed);
    dim3 grid(BATCH / TILE_S, SPLITS);   // 256 x 4 = 1024 blocks
    gbt_kernel<true><<<grid, BLOCK, 0, stream>>>(
        x, features, thresholds, packed, leaves, init_out, partials,
        TREES_PER_SPLIT, 0);
    const int tot = BATCH * N_CLASSES;
    gbt_reduce<<<(tot + 255) / 256, 256, 0, stream>>>(partials, init_out, out);
  } else if (ws_size >= partial_bytes) {
    float* partials = (float*)d_ws;
    dim3 grid(BATCH / TILE_S, SPLITS);
    gbt_kernel<false><<<grid, BLOCK, 0, stream>>>(
        x, features, thresholds, nullptr, leaves, init_out, partials,
        TREES_PER_SPLIT, 0);
    const int tot = BATCH * N_CLASSES;
    gbt_reduce<<<(tot + 255) / 256, 256, 0, stream>>>(partials, init_out, out);
  } else if (ws_size >= packed_bytes) {
    int2* packed = (int2*)d_ws;
    gbt_pack<<<(pack_elems + 255) / 256, 256, 0, stream>>>(features,
                                                           thresholds, packed);
    dim3 grid(BATCH / TILE_S, 1);
    gbt_kernel<true><<<grid, BLOCK, 0, stream>>>(
        x, features, thresholds, packed, leaves, init_out, out, N_TREES, 1);
  } else {
    dim3 grid(BATCH / TILE_S, 1);
    gbt_kernel<false><<<grid, BLOCK, 0, stream>>>(
        x, features, thresholds, nullptr, leaves, init_out, out, N_TREES, 1);
  }
}